// CapsNet_23905787969967
// MI455X (gfx1250) — compile-verified
//
#include <hip/hip_runtime.h>
#include <math.h>

typedef __attribute__((ext_vector_type(2))) float v2f;
typedef __attribute__((ext_vector_type(8))) float v8f;

// ---------------- workspace layout (floats) ----------------
// h    : conv1 output, 256*400                  @ [0, 102400)
// Bt   : im2col, col-major [48][20736]          @ [102400, 1097728)
// part : split-K partials, 8*256*36             @ [1097728, 1171456)
// caps : reduced caps conv out, 256*36          @ [1171456, 1180672)
#define WS_H     0
#define WS_BT    (256 * 400)
#define WS_PART  (WS_BT + 48 * 20736)
#define WS_CAPS  (WS_PART + 8 * 256 * 36)

// =====================================================================
// Kernel 1: conv1 (1 -> 256, 9x9, stride 1, VALID) + bias + ReLU
// =====================================================================
__global__ __launch_bounds__(256) void conv1_kernel(const float* __restrict__ x,
                                                    const float* __restrict__ w,
                                                    const float* __restrict__ b,
                                                    float* __restrict__ h) {
  int idx = blockIdx.x * 256 + threadIdx.x;   // exactly 102400 threads
  int c = idx / 400;
  int p = idx % 400;
  int oy = p / 20, ox = p % 20;
  const float* wc = w + c * 81;
  float acc = b[c];
#pragma unroll
  for (int ky = 0; ky < 9; ++ky) {
    const float* xr = x + (oy + ky) * 28 + ox;
#pragma unroll
    for (int kx = 0; kx < 9; ++kx)
      acc = fmaf(xr[kx], wc[ky * 9 + kx], acc);
  }
  h[idx] = fmaxf(acc, 0.0f);
}

// =====================================================================
// Kernel 2: materialize im2col column-major: Bt[n][k], n in [0,48), k in [0,20736)
//   Bt[n*20736 + k] = h[ic][oy*2+ky][ox*2+kx]  (n<36), else 0 (padding cols)
// One-time divisions here so the GEMM loop has none.
// =====================================================================
__global__ __launch_bounds__(256) void im2col_kernel(const float* __restrict__ h,
                                                     float* __restrict__ Bt) {
  int idx = blockIdx.x * 256 + threadIdx.x;   // exactly 48*20736 = 995328 threads
  int n = idx / 20736;
  int k = idx - n * 20736;
  float v = 0.0f;
  if (n < 36) {
    int ic = k / 81;  int r = k - ic * 81;  int ky = r / 9, kx = r - ky * 9;
    int oy = n / 6,   ox = n - oy * 6;
    v = h[ic * 400 + (oy * 2 + ky) * 20 + (ox * 2 + kx)];
  }
  Bt[idx] = v;
}

// =====================================================================
// Kernel 3: caps conv GEMM with fp32 WMMA, clean inner loop.
//   D[m][n] = sum_k A[m][k] * Bt[n][k]
// One wave per block. blockIdx.x = M-tile (16 rows), blockIdx.y = K-split (8).
// Per K-step: 4 coalesced b64 loads + 3 v_wmma_f32_16x16x4_f32. No divides,
// no branches -> EXEC stays all-ones as WMMA requires.
// =====================================================================
__global__ __launch_bounds__(32) void caps_gemm_kernel(const float* __restrict__ Aw,
                                                       const float* __restrict__ Bt,
                                                       float* __restrict__ part) {
  const int lane  = threadIdx.x;                 // 0..31
  const int mbase = blockIdx.x * 16;             // 16 M-tiles
  const int split = blockIdx.y;                  // 8 K-splits
  const int mrow  = mbase + (lane & 15);
  const int khalf = (lane >> 4) * 2;             // frag K sub-offset: 0 or 2
  const int n0    = (lane & 15);

  const float* __restrict__ arow = Aw + (size_t)mrow * 20736;
  const float* __restrict__ brow0 = Bt + (size_t)(n0)      * 20736;
  const float* __restrict__ brow1 = Bt + (size_t)(n0 + 16) * 20736;
  const float* __restrict__ brow2 = Bt + (size_t)(n0 + 32) * 20736;

  v8f acc0 = {}, acc1 = {}, acc2 = {};

  const int kbeg = split * 2592;                 // 20736 / 8
  const int kend = kbeg + 2592;

#pragma unroll 2
  for (int k = kbeg; k < kend; k += 4) {
    const int kk = k + khalf;
    v2f a  = *(const v2f*)(arow  + kk);
    v2f f0 = *(const v2f*)(brow0 + kk);
    v2f f1 = *(const v2f*)(brow1 + kk);
    v2f f2 = *(const v2f*)(brow2 + kk);
    __builtin_prefetch(arow + kk + 512, 0, 0);   // global_prefetch_b8 for A stream

    acc0 = __builtin_amdgcn_wmma_f32_16x16x4_f32(false, a, false, f0,
                                                 (short)0, acc0, false, false);
    acc1 = __builtin_amdgcn_wmma_f32_16x16x4_f32(false, a, false, f1,
                                                 (short)0, acc1, false, false);
    acc2 = __builtin_amdgcn_wmma_f32_16x16x4_f32(false, a, false, f2,
                                                 (short)0, acc2, false, false);
  }

  // Store partial tile. C/D layout: VGPR v -> M = v (lanes 0-15) / v+8 (lanes
  // 16-31), N = lane & 15.
  float* pout = part + (size_t)split * (256 * 36);
  const int moff = (lane >> 4) * 8;
  v8f accs[3] = {acc0, acc1, acc2};
#pragma unroll
  for (int t = 0; t < 3; ++t) {
    int n = t * 16 + n0;
    if (n < 36) {
#pragma unroll
      for (int v = 0; v < 8; ++v) {
        int m = mbase + v + moff;
        pout[m * 36 + n] = accs[t][v];
      }
    }
  }
}

// =====================================================================
// Kernel 4: reduce split-K partials + bias -> caps[256][36]
// =====================================================================
__global__ __launch_bounds__(256) void caps_reduce_kernel(const float* __restrict__ part,
                                                          const float* __restrict__ bias,
                                                          float* __restrict__ out) {
  int idx = blockIdx.x * 256 + threadIdx.x;      // exactly 9216 threads
  int m = idx / 36;
  float a = bias[m];
#pragma unroll
  for (int s = 0; s < 8; ++s) a += part[(size_t)s * 9216 + idx];
  out[idx] = a;
}

// =====================================================================
// Kernel 5: squash -> uhat -> 10 dynamic-routing iterations -> norms + b
// Single block, 384 threads (12 waves). Each thread owns 3 positions.
// Deterministic cross-wave reduction: ds_bpermute XOR butterfly + LDS.
// =====================================================================
__device__ __forceinline__ float lane_xor_f(float v, int mask) {
  int l = (int)(threadIdx.x & 31);
  int src = ((l ^ mask) << 2);
  return __int_as_float(__builtin_amdgcn_ds_bpermute(src, __float_as_int(v)));
}

__global__ __launch_bounds__(384) void routing_kernel(const float* __restrict__ caps,
                                                      const float* __restrict__ W,
                                                      const float* __restrict__ Wb,
                                                      float* __restrict__ out) {
  const int t    = threadIdx.x;
  const int lane = t & 31;
  const int wave = t >> 5;                       // 12 waves

  float uhat[3][16];
  float bb[3][10];

  // ---- squash(u) and uhat = u @ W^T + Wb, per position ----
#pragma unroll
  for (int s3 = 0; s3 < 3; ++s3) {
    int p = t + s3 * 384;                        // 0..1151 over (32,6,6)
    int i = p / 36;
    int rem = p - i * 36;                        // j*6+k
    float u[8];
    float nm2 = 0.0f;
#pragma unroll
    for (int d = 0; d < 8; ++d) {
      u[d] = caps[(i * 8 + d) * 36 + rem];
      nm2 = fmaf(u[d], u[d], nm2);
    }
    float scale = sqrtf(nm2) / (1.0f + nm2);
#pragma unroll
    for (int d = 0; d < 8; ++d) u[d] *= scale;
#pragma unroll
    for (int o = 0; o < 16; ++o) {
      float acc = Wb[p * 16 + o];
#pragma unroll
      for (int d = 0; d < 8; ++d)
        acc = fmaf(u[d], W[(p * 16 + o) * 8 + d], acc);
      uhat[s3][o] = acc;
    }
#pragma unroll
    for (int m = 0; m < 10; ++m) bb[s3][m] = 0.0f;
  }

  __shared__ float s_lds[10][16];
  __shared__ float wred[12][16];

  for (int it = 0; it < 10; ++it) {
    // ---- c = softmax(b) per position ----
    float c[3][10];
#pragma unroll
    for (int s3 = 0; s3 < 3; ++s3) {
      float mx = bb[s3][0];
#pragma unroll
      for (int m = 1; m < 10; ++m) mx = fmaxf(mx, bb[s3][m]);
      float sum = 0.0f;
#pragma unroll
      for (int m = 0; m < 10; ++m) {
        c[s3][m] = expf(bb[s3][m] - mx);
        sum += c[s3][m];
      }
      float inv = 1.0f / sum;
#pragma unroll
      for (int m = 0; m < 10; ++m) c[s3][m] *= inv;
    }

    // ---- s[m][o] = sum_p c[p][m]*uhat[p][o] ----
    for (int m = 0; m < 10; ++m) {
      float partv[16];
#pragma unroll
      for (int o = 0; o < 16; ++o) {
        float a = 0.0f;
#pragma unroll
        for (int s3 = 0; s3 < 3; ++s3) a = fmaf(c[s3][m], uhat[s3][o], a);
        partv[o] = a;
      }
      // intra-wave XOR butterfly (ds_bpermute_b32)
#pragma unroll
      for (int mask = 16; mask >= 1; mask >>= 1)
#pragma unroll
        for (int o = 0; o < 16; ++o)
          partv[o] += lane_xor_f(partv[o], mask);
      if (lane == 0)
#pragma unroll
        for (int o = 0; o < 16; ++o) wred[wave][o] = partv[o];
      __syncthreads();
      if (t < 16) {
        float a = 0.0f;
#pragma unroll
        for (int w = 0; w < 12; ++w) a += wred[w][t];
        s_lds[m][t] = a;
      }
      __syncthreads();
    }

    // ---- squash(s) per class ----
    if (t < 10) {
      float nm2 = 0.0f;
#pragma unroll
      for (int o = 0; o < 16; ++o) {
        float v = s_lds[t][o];
        nm2 = fmaf(v, v, nm2);
      }
      float scale = sqrtf(nm2) / (1.0f + nm2);
#pragma unroll
      for (int o = 0; o < 16; ++o) s_lds[t][o] *= scale;
    }
    __syncthreads();

    // ---- b[p][m] += sum_o s[m][o]*uhat[p][o] ----
#pragma unroll
    for (int s3 = 0; s3 < 3; ++s3)
#pragma unroll
      for (int m = 0; m < 10; ++m) {
        float a = 0.0f;
#pragma unroll
        for (int o = 0; o < 16; ++o) a = fmaf(s_lds[m][o], uhat[s3][o], a);
        bb[s3][m] += a;
      }

    // ---- last iteration: write class norms ----
    if (it == 9 && t < 10) {
      float nm2 = 0.0f;
#pragma unroll
      for (int o = 0; o < 16; ++o) {
        float v = s_lds[t][o];
        nm2 = fmaf(v, v, nm2);
      }
      out[t] = sqrtf(nm2);
    }
    __syncthreads();
  }

  // ---- write final b: out[10 + p*10 + m] ----
#pragma unroll
  for (int s3 = 0; s3 < 3; ++s3) {
    int p = t + s3 * 384;
#pragma unroll
    for (int m = 0; m < 10; ++m) out[10 + p * 10 + m] = bb[s3][m];
  }
}

// =====================================================================
extern "C" void kernel_launch(void* const* d_in, const int* in_sizes, int n_in,
                              void* d_out, int out_size, void* d_ws, size_t ws_size,
                              hipStream_t stream) {
  (void)in_sizes; (void)n_in; (void)out_size; (void)ws_size;
  const float* x       = (const float*)d_in[0];   // (1,1,28,28)
  const float* conv1_w = (const float*)d_in[1];   // (256,1,9,9)
  const float* conv1_b = (const float*)d_in[2];   // (256,)
  const float* caps_w  = (const float*)d_in[3];   // (256,256,9,9)
  const float* caps_b  = (const float*)d_in[4];   // (256,)
  const float* W       = (const float*)d_in[5];   // (32,6,6,16,8)
  const float* Wb      = (const float*)d_in[6];   // (32,6,6,16)
  float* out = (float*)d_out;                     // 10 norms + 11520 b

  float* ws   = (float*)d_ws;
  float* h    = ws + WS_H;                        // 256*400
  float* Bt   = ws + WS_BT;                       // 48*20736
  float* part = ws + WS_PART;                     // 8*256*36
  float* caps = ws + WS_CAPS;                     // 256*36

  conv1_kernel<<<400, 256, 0, stream>>>(x, conv1_w, conv1_b, h);
  im2col_kernel<<<3888, 256, 0, stream>>>(h, Bt);
  caps_gemm_kernel<<<dim3(16, 8), 32, 0, stream>>>(caps_w, Bt, part);
  caps_reduce_kernel<<<36, 256, 0, stream>>>(part, caps_b, caps);
  routing_kernel<<<1, 384, 0, stream>>>(caps, W, Wb, out);
}